// Rotation_85779086836290
// MI455X (gfx1250) — compile-verified
//
#include <hip/hip_runtime.h>
#include <math.h>

#ifndef M_PI
#define M_PI 3.14159265358979323846
#endif

#define N_SAMPLES 4096
#define DIM 12800

typedef __attribute__((ext_vector_type(2))) float v2f;
typedef __attribute__((ext_vector_type(8))) float v8f;

// ---- workspace layout (floats) ----
//  [0, 512)                      : J_l at (l-1)*128, row-major d*d
//  [512, 512 + 4096*4*192)       : per-(n,l) zero-padded D block, 16 rows x 12 cols
#define WS_JBASE   0
#define WS_JSTRIDE 128
#define WS_DBASE   512
#define WS_DSTRIDE 192

// ================= phase 0: compute J_l (device, double precision) ==========
__device__ double factd(int n) { double r = 1.0; for (int i = 2; i <= n; ++i) r *= (double)i; return r; }

__global__ void phase0_compute_J(float* __restrict__ ws) {
  int l = (int)threadIdx.x + 1;
  if (blockIdx.x != 0 || l > 4) return;
  const int d = 2 * l + 1;
  // Wigner small-d at beta = pi/2
  double dc[9][9];
  const double cb = cos(M_PI / 4.0), sb = sin(M_PI / 4.0);
  for (int i = 0; i < d; ++i) {
    int mp = i - l;
    for (int j = 0; j < d; ++j) {
      int m = j - l;
      double num = sqrt(factd(l + mp) * factd(l - mp) * factd(l + m) * factd(l - m));
      double tot = 0.0;
      int s0 = (m - mp) > 0 ? (m - mp) : 0;
      int s1 = (l + m) < (l - mp) ? (l + m) : (l - mp);
      for (int s = s0; s <= s1; ++s) {
        double den = factd(l + m - s) * factd(s) * factd(mp - m + s) * factd(l - mp - s);
        double sgn = ((mp - m + s) & 1) ? -1.0 : 1.0;
        tot += sgn * num / den * pow(cb, (double)(2 * l + m - mp - 2 * s)) * pow(sb, (double)(mp - m + 2 * s));
      }
      dc[i][j] = tot;
    }
  }
  // complex->real basis A
  double Are[9][9], Aim[9][9];
  for (int i = 0; i < 9; ++i) for (int j = 0; j < 9; ++j) { Are[i][j] = 0.0; Aim[i][j] = 0.0; }
  Are[l][l] = 1.0;
  const double is2 = 1.0 / sqrt(2.0);
  for (int m = 1; m <= l; ++m) {
    double sgn = (m & 1) ? -1.0 : 1.0;
    Are[l + m][l + m] = sgn * is2;
    Are[l + m][l - m] = is2;
    Aim[l - m][l + m] = -sgn * is2;  // -1j*(-1)^m/sqrt(2)
    Aim[l - m][l - m] = is2;         //  1j/sqrt(2)
  }
  // M = A @ dc (complex * real)
  double Mre[9][9], Mim[9][9];
  for (int i = 0; i < d; ++i)
    for (int j = 0; j < d; ++j) {
      double sr = 0.0, si = 0.0;
      for (int p = 0; p < d; ++p) { sr += Are[i][p] * dc[p][j]; si += Aim[i][p] * dc[p][j]; }
      Mre[i][j] = sr; Mim[i][j] = si;
    }
  // J = Re(M @ A^H)
  float* J = ws + WS_JBASE + (size_t)(l - 1) * WS_JSTRIDE;
  for (int i = 0; i < d; ++i)
    for (int j = 0; j < d; ++j) {
      double s = 0.0;
      for (int p = 0; p < d; ++p) s += Mre[i][p] * Are[j][p] + Mim[i][p] * Aim[j][p];
      J[i * d + j] = (float)s;
    }
}

// ================= phase 1: per-(n,l) D = Ea @ J @ Eb @ J^T @ Eg ============
template <int L>
__global__ void phase1_compute_D(const float* __restrict__ gamma,
                                 const float* __restrict__ beta,
                                 const float* __restrict__ alpha,
                                 float* __restrict__ ws) {
  constexpr int d = 2 * L + 1;
  __shared__ float Js[d * d];
  const float* Jg = ws + WS_JBASE + (size_t)(L - 1) * WS_JSTRIDE;
  for (int i = threadIdx.x; i < d * d; i += blockDim.x) Js[i] = Jg[i];
  __syncthreads();

  int n = blockIdx.x * blockDim.x + threadIdx.x;
  if (n >= N_SAMPLES) return;
  float g = gamma[n], b = beta[n], a = alpha[n];

  float cg[L + 1], sg[L + 1], cb[L + 1], sb[L + 1], ca[L + 1], sa[L + 1];
  cg[0] = 1.f; sg[0] = 0.f; cb[0] = 1.f; sb[0] = 0.f; ca[0] = 1.f; sa[0] = 0.f;
#pragma unroll
  for (int m = 1; m <= L; ++m) {
    sincosf((float)m * g, &sg[m], &cg[m]);
    sincosf((float)m * b, &sb[m], &cb[m]);
    sincosf((float)m * a, &sa[m], &ca[m]);
  }

  // T = J^T @ Eg : column rule T[:,j] = cos(m_j g)*J^T[:,j] + sin(m_j g)*J^T[:,d-1-j]
  float T[d * d];
#pragma unroll
  for (int j = 0; j < d; ++j) {
    int mj = j - L; int am = mj < 0 ? -mj : mj;
    float c = cg[am], s = (mj < 0 ? -sg[am] : sg[am]);
#pragma unroll
    for (int i = 0; i < d; ++i)
      T[i * d + j] = c * Js[j * d + i] + s * Js[(d - 1 - j) * d + i];
  }
  // T <- Eb @ T : row-pair rule (center row unchanged)
#pragma unroll
  for (int i = 0; i < L; ++i) {
    int k = d - 1 - i; float c = cb[L - i], s = sb[L - i];
#pragma unroll
    for (int j = 0; j < d; ++j) {
      float ti = T[i * d + j], tk = T[k * d + j];
      T[i * d + j] = c * ti + s * tk;
      T[k * d + j] = c * tk - s * ti;
    }
  }
  // V = J @ T
  float V[d * d];
#pragma unroll
  for (int i = 0; i < d; ++i)
#pragma unroll
    for (int j = 0; j < d; ++j) {
      float acc = 0.f;
#pragma unroll
      for (int p = 0; p < d; ++p) acc += Js[i * d + p] * T[p * d + j];
      V[i * d + j] = acc;
    }
  // V <- Ea @ V
#pragma unroll
  for (int i = 0; i < L; ++i) {
    int k = d - 1 - i; float c = ca[L - i], s = sa[L - i];
#pragma unroll
    for (int j = 0; j < d; ++j) {
      float vi = V[i * d + j], vk = V[k * d + j];
      V[i * d + j] = c * vi + s * vk;
      V[k * d + j] = c * vk - s * vi;
    }
  }
  // write zero-padded 16x12 block (row = output index i, col = k index)
  float* Dp = ws + WS_DBASE + (size_t)(n * 4 + (L - 1)) * WS_DSTRIDE;
#pragma unroll
  for (int r = 0; r < 16; ++r)
#pragma unroll
    for (int c2 = 0; c2 < 12; ++c2)
      Dp[r * 12 + c2] = (r < d && c2 < d) ? V[r * d + c2] : 0.0f;
}

// ================= phase 2: streaming WMMA rotation =========================
// Out(512 x d) = X(512 x d) @ D^T, done as 32 tiles of V_WMMA_F32_16X16X4_F32.
template <int L>
__device__ __forceinline__ void rotate_block(const float* __restrict__ x,
                                             const float* __restrict__ ws,
                                             float* __restrict__ out,
                                             int n, int lane) {
  constexpr int d = 2 * L + 1;
  constexpr int KC = (d + 3) / 4;          // K chunks of 4
  constexpr int OFF = 512 * L * L;         // l-block offset within a row
  const int col = lane & 15;
  const int hi = lane >> 4;                // 0: K = 0,1 ; 1: K = 2,3 (per chunk)

  // B operand: B[k, col] = D[col, k]; lane layout mirrors A (K0,K1 low lanes / K2,K3 high)
  const float* Dp = ws + WS_DBASE + (size_t)(n * 4 + (L - 1)) * WS_DSTRIDE;
  v2f Bv[KC];
#pragma unroll
  for (int c = 0; c < KC; ++c) {
    int kx = 4 * c + 2 * hi;
    Bv[c].x = Dp[col * 12 + kx];           // zero-padded, branch-free
    Bv[c].y = Dp[col * 12 + kx + 1];
  }

  const float* xb = x + (size_t)n * DIM + OFF;
  float* ob = out + (size_t)n * DIM + OFF;

#pragma unroll 1
  for (int t = 0; t < 32; ++t) {
    const float* xr = xb + (t * 16 + col) * d;   // A row = col (lane&15)
    v8f acc = {};
#pragma unroll
    for (int c = 0; c < KC; ++c) {
      int k0 = 4 * c + 2 * hi;
      v2f A;
      A.x = (k0 < d) ? xr[k0] : 0.0f;
      A.y = (k0 + 1 < d) ? xr[k0 + 1] : 0.0f;
      // D = A(16x4) x B(4x16) + C, fp32
      acc = __builtin_amdgcn_wmma_f32_16x16x4_f32(false, A, false, Bv[c],
                                                  (short)0, acc, false, false);
    }
    if (col < d) {                           // only first d output columns valid
      float* orow = ob + (size_t)(t * 16 + 8 * hi) * d + col;
#pragma unroll
      for (int r = 0; r < 8; ++r) orow[(size_t)r * d] = acc[r];
    }
  }
}

__device__ __forceinline__ void copy_l0(const float* __restrict__ x,
                                        float* __restrict__ out, int n, int lane) {
  const float4* src = (const float4*)(x + (size_t)n * DIM);
  float4* dst = (float4*)(out + (size_t)n * DIM);
#pragma unroll
  for (int i = 0; i < 4; ++i) dst[lane + 32 * i] = src[lane + 32 * i];  // 512 floats
}

__global__ void __launch_bounds__(160)
phase2_rotate(const float* __restrict__ x, const float* __restrict__ ws,
              float* __restrict__ out) {
  int n = blockIdx.x;
  int wave = (int)threadIdx.x >> 5;   // wave32
  int lane = (int)threadIdx.x & 31;
  switch (wave) {
    case 0: copy_l0(x, out, n, lane); break;
    case 1: rotate_block<1>(x, ws, out, n, lane); break;
    case 2: rotate_block<2>(x, ws, out, n, lane); break;
    case 3: rotate_block<3>(x, ws, out, n, lane); break;
    default: rotate_block<4>(x, ws, out, n, lane); break;
  }
}

// ============================================================================
extern "C" void kernel_launch(void* const* d_in, const int* in_sizes, int n_in,
                              void* d_out, int out_size, void* d_ws, size_t ws_size,
                              hipStream_t stream) {
  const float* gamma = (const float*)d_in[0];
  const float* beta  = (const float*)d_in[1];
  const float* alpha = (const float*)d_in[2];
  const float* x     = (const float*)d_in[3];
  float* out = (float*)d_out;
  float* ws  = (float*)d_ws;

  phase0_compute_J<<<1, 32, 0, stream>>>(ws);
  phase1_compute_D<1><<<16, 256, 0, stream>>>(gamma, beta, alpha, ws);
  phase1_compute_D<2><<<16, 256, 0, stream>>>(gamma, beta, alpha, ws);
  phase1_compute_D<3><<<16, 256, 0, stream>>>(gamma, beta, alpha, ws);
  phase1_compute_D<4><<<16, 256, 0, stream>>>(gamma, beta, alpha, ws);
  phase2_rotate<<<N_SAMPLES, 160, 0, stream>>>(x, ws, out);
}